// SCGP_LAYER_76751065579585
// MI455X (gfx1250) — compile-verified
//
#include <hip/hip_runtime.h>
#include <cstring>
#include <cstddef>

typedef __attribute__((ext_vector_type(2))) float v2f;
typedef __attribute__((ext_vector_type(8))) float v8f;

#define TILE 64
#define KC   32
#define APITCH (KC + 2)   // even pitch -> 8B-aligned v2f LDS loads

// ---------------------------------------------------------------------------
// Generic batched GEMM via V_WMMA_F32_16X16X4_F32.
// Logical: C[z][m][n] = epilogue( alpha * sum_k A[z][m][k]*B[z][k][n] )
// Strides express any transpose/reshape. Batch z decomposes as
// zo=z/innerB, zi=z%innerB with separate outer/inner strides per operand.
// All M,N multiples of 64, K multiples of 32; unit-stride operands are
// 16B aligned (all bases/strides are multiples of 64 floats).
// ---------------------------------------------------------------------------
struct GemmP {
  const float* A; int aRow, aCol, aBo, aBi;
  const float* B; int bRow, bCol, bBo, bBi;   // logical B[k][n]
  const float* Cin; int ciRow, ciCol, ciBo, ciBi;
  float* C; int cRow, cCol, cBo, cBi;
  const float* bias; int biasMode;            // 0 none, 1 per-col, 2 per-row
  const float* rn; const float* cn; int rnB, cnB; // RBF norms (per-batch stride)
  int Kdim;
  float alpha, beta, diag;                    // diag added where gm==gn
  int rbf;                                    // 1: C=exp(-0.5*max(rn+cn-2acc,0))
  int innerB;
};

__global__ __launch_bounds__(128)
void gemm_wmma_kernel(GemmP p) {
  __shared__ float As[TILE][APITCH];
  __shared__ float Bs[TILE][APITCH];          // stored transposed: Bs[n][k]

  const int z  = blockIdx.z;
  const int zo = z / p.innerB, zi = z % p.innerB;
  const float* Ag = p.A + (size_t)zo * p.aBo + (size_t)zi * p.aBi;
  const float* Bg = p.B + (size_t)zo * p.bBo + (size_t)zi * p.bBi;

  const int m0 = blockIdx.x * TILE;
  const int n0 = blockIdx.y * TILE;
  const int tid  = threadIdx.x;
  const int lane = tid & 31;
  const int wid  = tid >> 5;
  const int wm = (wid >> 1) * 32;             // wave 32x32 quadrant
  const int wn = (wid & 1) * 32;
  const int lh = (lane >> 4) * 2;             // K sub-offset 0 or 2
  const int lm = lane & 15;

  v8f acc[2][2];
  const v8f vzero = {0.f,0.f,0.f,0.f,0.f,0.f,0.f,0.f};
  acc[0][0] = vzero; acc[0][1] = vzero; acc[1][0] = vzero; acc[1][1] = vzero;

  for (int k0 = 0; k0 < p.Kdim; k0 += KC) {
    __syncthreads();
    // ---- Stage A tile [64][KC] (float4 fast paths -> global_load_b128)
    if (p.aCol == 1) {                 // contiguous along K
      for (int t = tid; t < TILE * (KC / 4); t += 128) {
        int r = t >> 3, c4 = (t & 7) << 2;
        float4 vv = *(const float4*)&Ag[(size_t)(m0 + r) * p.aRow + (k0 + c4)];
        As[r][c4+0] = vv.x; As[r][c4+1] = vv.y; As[r][c4+2] = vv.z; As[r][c4+3] = vv.w;
      }
    } else if (p.aRow == 1) {          // contiguous along M (transposed operand)
      for (int t = tid; t < (TILE / 4) * KC; t += 128) {
        int c = t >> 4, r4 = (t & 15) << 2;
        float4 vv = *(const float4*)&Ag[(size_t)(m0 + r4) + (size_t)(k0 + c) * p.aCol];
        As[r4+0][c] = vv.x; As[r4+1][c] = vv.y; As[r4+2][c] = vv.z; As[r4+3][c] = vv.w;
      }
    } else {
      for (int t = tid; t < TILE * KC; t += 128) {
        int r = t >> 5, c = t & (KC - 1);
        As[r][c] = Ag[(size_t)(m0 + r) * p.aRow + (size_t)(k0 + c) * p.aCol];
      }
    }
    // ---- Stage B tile transposed: Bs[n][k]
    if (p.bRow == 1) {                 // contiguous along K
      for (int t = tid; t < TILE * (KC / 4); t += 128) {
        int r = t >> 3, c4 = (t & 7) << 2;
        float4 vv = *(const float4*)&Bg[(size_t)(k0 + c4) + (size_t)(n0 + r) * p.bCol];
        Bs[r][c4+0] = vv.x; Bs[r][c4+1] = vv.y; Bs[r][c4+2] = vv.z; Bs[r][c4+3] = vv.w;
      }
    } else if (p.bCol == 1) {          // contiguous along N
      for (int t = tid; t < (TILE / 4) * KC; t += 128) {
        int c = t >> 4, n4 = (t & 15) << 2;
        float4 vv = *(const float4*)&Bg[(size_t)(k0 + c) * p.bRow + (n0 + n4)];
        Bs[n4+0][c] = vv.x; Bs[n4+1][c] = vv.y; Bs[n4+2][c] = vv.z; Bs[n4+3][c] = vv.w;
      }
    } else {
      for (int t = tid; t < TILE * KC; t += 128) {
        int r = t >> 5, c = t & (KC - 1);
        Bs[r][c] = Bg[(size_t)(k0 + c) * p.bRow + (size_t)(n0 + r) * p.bCol];
      }
    }
    if (k0 + KC < p.Kdim) {  // prefetch next chunk (global_prefetch_b8)
      __builtin_prefetch(&Ag[(size_t)(m0 + (tid >> 1)) * p.aRow + (size_t)(k0 + KC) * p.aCol], 0, 1);
      __builtin_prefetch(&Bg[(size_t)(k0 + KC) * p.bRow + (size_t)(n0 + (tid >> 1)) * p.bCol], 0, 1);
    }
    __syncthreads();

#pragma unroll
    for (int kk = 0; kk < KC; kk += 4) {
      v2f a0 = *(const v2f*)&As[wm + lm     ][kk + lh];
      v2f a1 = *(const v2f*)&As[wm + 16 + lm][kk + lh];
      v2f b0 = *(const v2f*)&Bs[wn + lm     ][kk + lh];
      v2f b1 = *(const v2f*)&Bs[wn + 16 + lm][kk + lh];
      acc[0][0] = __builtin_amdgcn_wmma_f32_16x16x4_f32(false, a0, false, b0, (short)0, acc[0][0], false, false);
      acc[0][1] = __builtin_amdgcn_wmma_f32_16x16x4_f32(false, a0, false, b1, (short)0, acc[0][1], false, false);
      acc[1][0] = __builtin_amdgcn_wmma_f32_16x16x4_f32(false, a1, false, b0, (short)0, acc[1][0], false, false);
      acc[1][1] = __builtin_amdgcn_wmma_f32_16x16x4_f32(false, a1, false, b1, (short)0, acc[1][1], false, false);
    }
  }

  // Epilogue. C/D layout: VGPR r -> M=r (lanes 0-15) / M=8+r (lanes 16-31), N=lane&15
  const size_t cBase  = (size_t)zo * p.cBo  + (size_t)zi * p.cBi;
  const size_t ciBase = (size_t)zo * p.ciBo + (size_t)zi * p.ciBi;
  const int rowOff = (lane < 16) ? 0 : 8;
  const size_t rnBase = (size_t)z * p.rnB;
  const size_t cnBase = (size_t)z * p.cnB;

#pragma unroll
  for (int ti = 0; ti < 2; ++ti)
#pragma unroll
    for (int tj = 0; tj < 2; ++tj)
#pragma unroll
      for (int r = 0; r < 8; ++r) {
        int gm = m0 + wm + ti * 16 + rowOff + r;
        int gn = n0 + wn + tj * 16 + lm;
        float v = p.alpha * acc[ti][tj][r];
        if (p.rbf) {
          float d2 = p.rn[rnBase + gm] + p.cn[cnBase + gn] - 2.0f * v;
          v = expf(-0.5f * fmaxf(d2, 0.0f));
        } else {
          if (p.beta != 0.0f)
            v += p.beta * p.Cin[ciBase + (size_t)gm * p.ciRow + (size_t)gn * p.ciCol];
          if (p.bias)
            v += (p.biasMode == 1) ? p.bias[gn] : p.bias[gm];
          if (p.diag != 0.0f && gm == gn) v += p.diag;
        }
        p.C[cBase + (size_t)gm * p.cRow + (size_t)gn * p.cCol] = v;
      }
}

// ---------------------------------------------------------------------------
// Row squared-norms: N[r] = sum_v T[r*V+v]^2
// ---------------------------------------------------------------------------
__global__ void rownorm_kernel(const float* __restrict__ T, float* __restrict__ N,
                               int totalRows, int V) {
  int r = blockIdx.x * blockDim.x + threadIdx.x;
  if (r >= totalRows) return;
  const float* p = T + (size_t)r * V;
  float s = 0.f;
  for (int i = 0; i < V; i += 4) {
    float4 x = *(const float4*)&p[i];
    s += x.x*x.x + x.y*x.y + x.z*x.z + x.w*x.w;
  }
  N[r] = s;
}

// O = a*X + b*Y + d*I  (batched n x n)
__global__ void mmcomb_kernel(const float* __restrict__ X, const float* __restrict__ Y,
                              float* __restrict__ O, float a, float b, float d, int n) {
  int per = n * n;
  int i = blockIdx.x * blockDim.x + threadIdx.x;
  if (i >= per) return;
  size_t base = (size_t)blockIdx.y * per;
  float v = a * X[base + i];
  if (Y) v += b * Y[base + i];
  if (d != 0.f && (i / n == i % n)) v += d;
  O[base + i] = v;
}

// Batched 64x64 Gauss-Jordan inverse in LDS (SPD + diag-loaded inputs).
__global__ __launch_bounds__(256)
void inv64_kernel(const float* __restrict__ E, float* __restrict__ O, int bs) {
  __shared__ float A[64][65];
  __shared__ float Inv[64][65];
  __shared__ float pinv;
  const float* Eb = E + (size_t)blockIdx.x * bs;
  float* Ob = O + (size_t)blockIdx.x * bs;
  int row = threadIdx.x & 63, seg = threadIdx.x >> 6, j0 = seg * 16;
  for (int j = j0; j < j0 + 16; ++j) {
    A[row][j] = Eb[row * 64 + j];
    Inv[row][j] = (row == j) ? 1.f : 0.f;
  }
  __syncthreads();
  for (int k = 0; k < 64; ++k) {
    if (threadIdx.x == 0) pinv = 1.0f / A[k][k];
    __syncthreads();
    float f = A[row][k] * pinv;
    __syncthreads();
    if (row != k)
      for (int j = j0; j < j0 + 16; ++j) {
        A[row][j]   -= f * A[k][j];
        Inv[row][j] -= f * Inv[k][j];
      }
    __syncthreads();
    if (row == k)
      for (int j = j0; j < j0 + 16; ++j) { A[k][j] *= pinv; Inv[k][j] *= pinv; }
    __syncthreads();
  }
  for (int j = j0; j < j0 + 16; ++j) Ob[row * 64 + j] = Inv[row][j];
}

// ---------------------------------------------------------------------------
// Blocked Cholesky, panel step (NB=64): factor diagonal block in LDS,
// build L11^{-1} in LDS, TRSM the panel rows in place (increasing-j order
// makes in-place safe). Trailing SYRK is done by gemm_wmma_kernel launches.
// ---------------------------------------------------------------------------
__global__ __launch_bounds__(256)
void chol_panel_kernel(float* __restrict__ Ab, int L, int bStride, int kb) {
  float* A = Ab + (size_t)blockIdx.x * bStride;
  __shared__ float T[64][65];     // diagonal block (lower)
  __shared__ float Tin[64][65];   // L11^{-1} (lower)
  __shared__ float dinv[64];
  const int tid = threadIdx.x;

  for (int t = tid; t < 64 * 64; t += 256) {
    int i = t >> 6, j = t & 63;
    T[i][j] = A[(size_t)(kb + i) * L + kb + j];
  }
  __syncthreads();
  // Factor 64x64 in LDS (right-looking rank-1, all in LDS)
  for (int k = 0; k < 64; ++k) {
    if (tid == 0) {
      float d = sqrtf(fmaxf(T[k][k], 1e-20f));
      T[k][k] = d;
      dinv[k] = 1.0f / d;
    }
    __syncthreads();
    for (int i = k + 1 + tid; i < 64; i += 256) T[i][k] *= dinv[k];
    __syncthreads();
    for (int i = k + 1 + tid; i < 64; i += 256) {
      float c = T[i][k];
      for (int j = k + 1; j <= i; ++j) T[i][j] -= c * T[j][k];
    }
    __syncthreads();
  }
  // Write back diagonal block (zero upper within block)
  for (int t = tid; t < 64 * 64; t += 256) {
    int i = t >> 6, j = t & 63;
    A[(size_t)(kb + i) * L + kb + j] = (j <= i) ? T[i][j] : 0.f;
  }
  // Triangular inverse L11^{-1}: thread c solves L11 x = e_c (forward subst.)
  if (tid < 64) {
    int c = tid;
    for (int j = 0; j < 64; ++j) {
      if (j < c) { Tin[j][c] = 0.f; continue; }
      float s = (j == c) ? 1.f : 0.f;
      for (int t = c; t < j; ++t) s -= T[j][t] * Tin[t][c];
      Tin[j][c] = s * dinv[j];
    }
  }
  __syncthreads();
  // Panel TRSM in place: L21[i][j] = sum_{t>=j} A21[i][t] * L11^{-1}[t][j]... row-major:
  // L21 = A21 * L11^{-1}; computing j ascending reads only row[t>=j], so in place is safe.
  const int nrows = L - kb - 64;
  for (int i = tid; i < nrows; i += 256) {
    float* row = A + (size_t)(kb + 64 + i) * L + kb;
    for (int j = 0; j < 64; ++j) {
      float s = 0.f;
      for (int t = j; t < 64; ++t) s += row[t] * Tin[t][j];
      row[j] = s;
    }
  }
}

__global__ void zero_upper_kernel(float* __restrict__ Ab, int L, int bStride) {
  float* A = Ab + (size_t)blockIdx.x * bStride;
  for (int idx = threadIdx.x + blockIdx.y * blockDim.x; idx < L * L;
       idx += blockDim.x * gridDim.y) {
    int i = idx / L, j = idx % L;
    if (j > i) A[idx] = 0.f;
  }
}

__global__ void set_scalar_kernel(float* p, float v) {
  if (threadIdx.x == 0 && blockIdx.x == 0) *p = v;
}

// log_joint traces: (1/n4)[tr(Aq T0)+tr(Ak T0)] + tr(Kinv2 Cq) + tr(Kinv2 Ck)
__global__ __launch_bounds__(256)
void logjoint_kernel(const float* Aq, const float* Ak, const float* T0,
                     const float* Kinv2, const float* Cq, const float* Ck,
                     float* out, float inv_n4, float scale) {
  __shared__ float red[256];
  size_t base = (size_t)blockIdx.x * 4096;
  float s = 0.f;
  for (int t = threadIdx.x; t < 4096; t += 256) {
    int i = t >> 6, j = t & 63;
    float tij = T0[base + j * 64 + i];
    float x = (Aq[base + i * 64 + j] + Ak[base + i * 64 + j]) * tij;
    float y = Kinv2[base + i * 64 + j] * (Cq[base + j * 64 + i] + Ck[base + j * 64 + i]);
    s += inv_n4 * x + y;
  }
  red[threadIdx.x] = s;
  __syncthreads();
  for (int st = 128; st > 0; st >>= 1) {
    if (threadIdx.x < st) red[threadIdx.x] += red[threadIdx.x + st];
    __syncthreads();
  }
  if (threadIdx.x == 0) atomicAdd(out, red[0] * scale);
}

// ---------------------------------------------------------------------------
// Host orchestration
// ---------------------------------------------------------------------------
static GemmP mk() { GemmP p; memset(&p, 0, sizeof(p)); p.innerB = 1; p.alpha = 1.f; return p; }

static inline void launch_gemm(hipStream_t st, const GemmP& p, int M, int N, int batch) {
  dim3 grid(M / TILE, N / TILE, batch);
  gemm_wmma_kernel<<<grid, dim3(128), 0, st>>>(p);
}

extern "C" void kernel_launch(void* const* d_in, const int* in_sizes, int n_in,
                              void* d_out, int out_size, void* d_ws, size_t ws_size,
                              hipStream_t stream) {
  (void)in_sizes; (void)n_in; (void)out_size; (void)ws_size;
  const int Bb = 8, Hh = 8, Ll = 512, Dd = 512, Vv = 64, Mm = 64, BH = 64;
  const float n2 = 1.0f, inv_n4 = 1.0f, JIT = 1e-5f;

  const float* x    = (const float*)d_in[0];
  const float* Wq   = (const float*)d_in[1];
  const float* Wk   = (const float*)d_in[2];
  const float* Wv   = (const float*)d_in[3];
  const float* Wx0  = (const float*)d_in[4];
  const float* Wind = (const float*)d_in[5];
  const float* bind = (const float*)d_in[6];
  const float* Wo   = (const float*)d_in[7];
  const float* bo   = (const float*)d_in[8];
  const float* eps  = (const float*)d_in[9];
  float* out = (float*)d_out;

  const size_t szLV = (size_t)BH * Ll * Vv;
  const size_t szMV = (size_t)BH * Mm * Vv;
  const size_t szLM = (size_t)BH * Ll * Mm;
  const size_t szMM = (size_t)BH * Mm * Mm;
  const size_t szLL = (size_t)BH * Ll * Ll;

  size_t off = 0;
  auto alloc = [&](size_t n) { float* p = (float*)d_ws + off; off += n; return p; };
  float* q    = alloc(szLV);
  float* kbuf = alloc(szLV);
  float* vbuf = alloc(szLV);
  float* x0   = alloc(szLV);
  float* xm   = alloc(szMV);
  float* nq   = alloc((size_t)BH * Ll);
  float* nk   = alloc((size_t)BH * Ll);
  float* n0   = alloc((size_t)BH * Ll);
  float* nxm  = alloc((size_t)BH * Mm);
  float* Kqm  = alloc(szLM);
  float* K0m  = alloc(szLM);
  float* Kkl  = alloc(szLM);
  float* Kmm  = alloc(szMM);
  float* K0   = alloc(szLL);        // reused as covar/chol matrix after T0
  float* Pb   = alloc(szLM);
  float* T0   = alloc(szMM);
  float* Gm   = alloc(szMM);
  float* Ckm  = alloc(szMM);
  float* Cqm  = alloc(szMM);
  float* Emm  = alloc(szMM);
  float* Ell  = alloc(szMM);
  float* E2   = alloc(szMM);
  float* KmmI = alloc(szMM);
  float* KllI = alloc(szMM);
  float* Kin2 = alloc(szMM);
  float* Zv   = alloc(szMM);
  float* Az   = alloc(szMM);
  float* P2   = alloc(szMM);
  float* P3   = alloc(szMM);
  float* inl  = alloc(szMM);
  float* Hm1  = alloc(szMM);
  float* Hm3  = alloc(szMM);
  float* Hm4  = alloc(szMM);
  float* inm  = alloc(szMM);
  float* mean = alloc(szLV);
  float* Wb   = alloc(szLM);
  float* smp  = alloc((size_t)Bb * Ll * Dd);
  float* Tq   = alloc(szMM);
  float* Aqm  = alloc(szMM);
  float* Tk   = alloc(szMM);
  float* Akm  = alloc(szMM);
  float* C1 = K0;                   // alias (used after K0 is consumed)

  // ---- 1) Head projections: q/k/v/x0[b,h,l,v] = sum_d x[b,l,d]*W[h*64+v,d]
  {
    const float* Ws[4] = { Wq, Wk, Wv, Wx0 };
    float* Os[4] = { q, kbuf, vbuf, x0 };
    for (int i = 0; i < 4; ++i) {
      GemmP p = mk(); p.innerB = Hh;
      p.A = x;     p.aRow = Dd; p.aCol = 1;  p.aBo = Ll * Dd; p.aBi = 0;
      p.B = Ws[i]; p.bRow = 1;  p.bCol = Dd; p.bBo = 0;       p.bBi = Vv * Dd;
      p.C = Os[i]; p.cRow = Vv; p.cCol = 1;  p.cBo = Hh * Ll * Vv; p.cBi = Ll * Vv;
      p.Kdim = Dd;
      launch_gemm(stream, p, Ll, Vv, BH);
    }
  }
  // ---- 2) Inducing points xm[b,h,m,v] = sum_l x[b,l,h*64+v]*Wind[m,l] + bind[m]
  {
    GemmP p = mk(); p.innerB = Hh;
    p.A = Wind; p.aRow = Ll; p.aCol = 1; p.aBo = 0; p.aBi = 0;
    p.B = x;    p.bRow = Dd; p.bCol = 1; p.bBo = Ll * Dd; p.bBi = Vv;
    p.C = xm;   p.cRow = Vv; p.cCol = 1; p.cBo = Hh * Mm * Vv; p.cBi = Mm * Vv;
    p.bias = bind; p.biasMode = 2;
    p.Kdim = Ll;
    launch_gemm(stream, p, Mm, Vv, BH);
  }
  // ---- 3) Squared norms
  rownorm_kernel<<<(BH * Ll + 255) / 256, 256, 0, stream>>>(q,    nq,  BH * Ll, Vv);
  rownorm_kernel<<<(BH * Ll + 255) / 256, 256, 0, stream>>>(kbuf, nk,  BH * Ll, Vv);
  rownorm_kernel<<<(BH * Ll + 255) / 256, 256, 0, stream>>>(x0,   n0,  BH * Ll, Vv);
  rownorm_kernel<<<(BH * Mm + 255) / 256, 256, 0, stream>>>(xm,   nxm, BH * Mm, Vv);

  // ---- 4) RBF kernel matrices
  auto rbf_LM = [&](const float* Abuf, const float* rnv, float* Cbuf) {
    GemmP p = mk();
    p.A = Abuf; p.aRow = Vv; p.aCol = 1; p.aBo = Ll * Vv;
    p.B = xm;   p.bRow = 1;  p.bCol = Vv; p.bBo = Mm * Vv;   // B[k=v][n=m]=xm[m,v]
    p.C = Cbuf; p.cRow = Mm; p.cCol = 1; p.cBo = Ll * Mm;
    p.rbf = 1; p.rn = rnv; p.rnB = Ll; p.cn = nxm; p.cnB = Mm;
    p.Kdim = Vv;
    launch_gemm(stream, p, Ll, Mm, BH);
  };
  rbf_LM(q,    nq, Kqm);
  rbf_LM(x0,   n0, K0m);
  rbf_LM(kbuf, nk, Kkl);
  {
    GemmP p = mk();                            // K_mm = rbf(xm, xm)
    p.A = xm; p.aRow = Vv; p.aCol = 1; p.aBo = Mm * Vv;
    p.B = xm; p.bRow = 1;  p.bCol = Vv; p.bBo = Mm * Vv;
    p.C = Kmm; p.cRow = Mm; p.cCol = 1; p.cBo = Mm * Mm;
    p.rbf = 1; p.rn = nxm; p.rnB = Mm; p.cn = nxm; p.cnB = Mm;
    p.Kdim = Vv;
    launch_gemm(stream, p, Mm, Mm, BH);
  }
  {
    GemmP p = mk();                            // K_0 = rbf(x0, x0)  [L x L]
    p.A = x0; p.aRow = Vv; p.aCol = 1; p.aBo = Ll * Vv;
    p.B = x0; p.bRow = 1;  p.bCol = Vv; p.bBo = Ll * Vv;
    p.C = K0; p.cRow = Ll; p.cCol = 1; p.cBo = Ll * Ll;
    p.rbf = 1; p.rn = n0; p.rnB = Ll; p.cn = n0; p.cnB = Ll;
    p.Kdim = Vv;
    launch_gemm(stream, p, Ll, Ll, BH);
  }

  // ---- 5) T0 = K_0m^T @ K_0 @ K_0m (consumes K0; K0 buffer is then free)
  {
    GemmP p = mk();                            // P = K_0 @ K_0m
    p.A = K0;  p.aRow = Ll; p.aCol = 1; p.aBo = Ll * Ll;
    p.B = K0m; p.bRow = Mm; p.bCol = 1; p.bBo = Ll * Mm;
    p.C = Pb;  p.cRow = Mm; p.cCol = 1; p.cBo = Ll * Mm;
    p.Kdim = Ll;
    launch_gemm(stream, p, Ll, Mm, BH);
  }
  {
    GemmP p = mk();                            // T0 = K_0m^T @ P
    p.A = K0m; p.aRow = 1;  p.aCol = Mm; p.aBo = Ll * Mm;
    p.B = Pb;  p.bRow = Mm; p.bCol = 1;  p.bBo = Ll * Mm;
    p.C = T0;  p.cRow = Mm; p.cCol = 1;  p.cBo = Mm * Mm;
    p.Kdim = Ll;
    launch_gemm(stream, p, Mm, Mm, BH);
  }

  // ---- 6) Gram matrices G=K_0m^T K_0m, Ck=K_kl^T K_kl, Cq=K_qm^T K_qm
  auto gram = [&](const float* Xb, float* Cb) {
    GemmP p = mk();
    p.A = Xb; p.aRow = 1;  p.aCol = Mm; p.aBo = Ll * Mm;
    p.B = Xb; p.bRow = Mm; p.bCol = 1;  p.bBo = Ll * Mm;
    p.C = Cb; p.cRow = Mm; p.cCol = 1;  p.cBo = Mm * Mm;
    p.Kdim = Ll;
    launch_gemm(stream, p, Mm, Mm, BH);
  };
  gram(K0m, Gm);
  gram(Kkl, Ckm);
  gram(Kqm, Cqm);

  // ---- 7) Inverses: Kmm_inv=inv(Kmm+G), Kll_inv=inv(Kmm+Ck), Kinv2=inv(Kmm+I)
  dim3 cg(16, BH);
  mmcomb_kernel<<<cg, 256, 0, stream>>>(Kmm, Gm,  Emm, 1.f, 1.f / n2, 0.f, Mm);
  mmcomb_kernel<<<cg, 256, 0, stream>>>(Kmm, Ckm, Ell, 1.f, 1.f / n2, 0.f, Mm);
  mmcomb_kernel<<<cg, 256, 0, stream>>>(Kmm, nullptr, E2, 1.f, 0.f, 1.f / n2, Mm);
  inv64_kernel<<<BH, 256, 0, stream>>>(Emm, KmmI, Mm * Mm);
  inv64_kernel<<<BH, 256, 0, stream>>>(Ell, KllI, Mm * Mm);
  inv64_kernel<<<BH, 256, 0, stream>>>(E2,  Kin2, Mm * Mm);

  auto gemmMM = [&](const float* Ab, const float* Bbuf, float* Cb,
                    float alpha, float beta, const float* Cinb) {
    GemmP p = mk();
    p.A = Ab;   p.aRow = Mm; p.aCol = 1; p.aBo = Mm * Mm;
    p.B = Bbuf; p.bRow = Mm; p.bCol = 1; p.bBo = Mm * Mm;
    p.C = Cb;   p.cRow = Mm; p.cCol = 1; p.cBo = Mm * Mm;
    p.alpha = alpha; p.beta = beta;
    if (Cinb) { p.Cin = Cinb; p.ciRow = Mm; p.ciCol = 1; p.ciBo = Mm * Mm; }
    p.Kdim = Mm;
    launch_gemm(stream, p, Mm, Mm, BH);
  };

  // ---- 8) mean path: Zv = K_kl^T v ; Az = Kll_inv Zv ; P2 = G Az ; P3 = Kmm_inv P2
  {
    GemmP p = mk();                            // Zv = K_lk @ v
    p.A = Kkl;  p.aRow = 1;  p.aCol = Mm; p.aBo = Ll * Mm;
    p.B = vbuf; p.bRow = Vv; p.bCol = 1;  p.bBo = Ll * Vv;
    p.C = Zv;   p.cRow = Vv; p.cCol = 1;  p.cBo = Mm * Vv;
    p.Kdim = Ll;
    launch_gemm(stream, p, Mm, Vv, BH);
  }
  gemmMM(KllI, Zv, Az, 1.f, 0.f, nullptr);
  gemmMM(Gm,   Az, P2, 1.f, 0.f, nullptr);
  gemmMM(KmmI, P2, P3, 1.f, 0.f, nullptr);
  {
    GemmP p = mk();                            // mean = (1/n4) K_qm @ P3
    p.A = Kqm; p.aRow = Mm; p.aCol = 1; p.aBo = Ll * Mm;
    p.B = P3;  p.bRow = Vv; p.bCol = 1; p.bBo = Mm * Vv;
    p.C = mean; p.cRow = Vv; p.cCol = 1; p.cBo = Ll * Vv;
    p.alpha = inv_n4;
    p.Kdim = Mm;
    launch_gemm(stream, p, Ll, Vv, BH);
  }

  // ---- 9) inner_l = Kll_inv + (1/n4) Az Az^T
  {
    GemmP p = mk();
    p.A = Az; p.aRow = Vv; p.aCol = 1;  p.aBo = Mm * Vv;
    p.B = Az; p.bRow = 1;  p.bCol = Vv; p.bBo = Mm * Vv;   // B[k=v][n]=Az[n,v]
    p.C = inl; p.cRow = Mm; p.cCol = 1; p.cBo = Mm * Mm;
    p.alpha = inv_n4; p.beta = 1.f;
    p.Cin = KllI; p.ciRow = Mm; p.ciCol = 1; p.ciBo = Mm * Mm;
    p.Kdim = Vv;
    launch_gemm(stream, p, Mm, Mm, BH);
  }
  // ---- 10) inner_m = Kmm_inv + (1/n4) Kmm_inv (n2 G + G inner_l G) Kmm_inv
  gemmMM(inl, Gm, Hm1, 1.f, 0.f, nullptr);          // H1 = inner_l G
  gemmMM(Gm, Hm1, Hm3, 1.f, n2, Gm);                // H3 = G H1 + n2*G
  gemmMM(KmmI, Hm3, Hm4, 1.f, 0.f, nullptr);        // H4 = Kmm_inv H3
  gemmMM(Hm4, KmmI, inm, inv_n4, 1.f, KmmI);        // inner_m

  // ---- 11) covar = n2 I + K_qm inner_m K_qm^T - mean mean^T + JITTER I  (into C1)
  {
    GemmP p = mk();                            // Wb = K_qm @ inner_m
    p.A = Kqm; p.aRow = Mm; p.aCol = 1; p.aBo = Ll * Mm;
    p.B = inm; p.bRow = Mm; p.bCol = 1; p.bBo = Mm * Mm;
    p.C = Wb;  p.cRow = Mm; p.cCol = 1; p.cBo = Ll * Mm;
    p.Kdim = Mm;
    launch_gemm(stream, p, Ll, Mm, BH);
  }
  {
    GemmP p = mk();                            // C1 = Wb @ K_qm^T + (n2+JIT) I
    p.A = Wb;  p.aRow = Mm; p.aCol = 1;  p.aBo = Ll * Mm;
    p.B = Kqm; p.bRow = 1;  p.bCol = Mm; p.bBo = Ll * Mm;
    p.C = C1;  p.cRow = Ll; p.cCol = 1;  p.cBo = Ll * Ll;
    p.diag = n2 + JIT;
    p.Kdim = Mm;
    launch_gemm(stream, p, Ll, Ll, BH);
  }
  {
    GemmP p = mk();                            // C1 += -mean @ mean^T
    p.A = mean; p.aRow = Vv; p.aCol = 1;  p.aBo = Ll * Vv;
    p.B = mean; p.bRow = 1;  p.bCol = Vv; p.bBo = Ll * Vv;
    p.C = C1;   p.cRow = Ll; p.cCol = 1;  p.cBo = Ll * Ll;
    p.alpha = -1.f; p.beta = 1.f;
    p.Cin = C1; p.ciRow = Ll; p.ciCol = 1; p.ciBo = Ll * Ll;
    p.Kdim = Vv;
    launch_gemm(stream, p, Ll, Ll, BH);
  }

  // ---- 12) Blocked Cholesky: panel (LDS factor + TRSM) then WMMA SYRK update
  for (int kb = 0; kb < Ll / 64; ++kb) {
    chol_panel_kernel<<<BH, 256, 0, stream>>>(C1, Ll, Ll * Ll, kb * 64);
    int trail = Ll - (kb + 1) * 64;
    if (trail > 0) {
      float* panel = C1 + (size_t)(kb + 1) * 64 * Ll + (size_t)kb * 64;
      float* cbase = C1 + (size_t)(kb + 1) * 64 * Ll + (size_t)(kb + 1) * 64;
      GemmP p = mk();
      p.A = panel; p.aRow = Ll; p.aCol = 1;  p.aBo = Ll * Ll;   // A21 [trail x 64]
      p.B = panel; p.bRow = 1;  p.bCol = Ll; p.bBo = Ll * Ll;   // A21^T
      p.C = cbase; p.cRow = Ll; p.cCol = 1;  p.cBo = Ll * Ll;
      p.Cin = cbase; p.ciRow = Ll; p.ciCol = 1; p.ciBo = Ll * Ll;
      p.alpha = -1.f; p.beta = 1.f;
      p.Kdim = 64;
      launch_gemm(stream, p, trail, trail, BH);
    }
  }
  zero_upper_kernel<<<dim3(BH, 32), 256, 0, stream>>>(C1, Ll, Ll * Ll);

  // ---- 13) samples: smp[b,l,h*64+v] = mean + chol @ eps
  {
    GemmP p = mk(); p.innerB = Hh;
    p.A = C1;  p.aRow = Ll; p.aCol = 1; p.aBo = Hh * Ll * Ll; p.aBi = Ll * Ll;
    p.B = eps; p.bRow = Vv; p.bCol = 1; p.bBo = Hh * Ll * Vv; p.bBi = Ll * Vv;
    p.C = smp; p.cRow = Dd; p.cCol = 1; p.cBo = Ll * Dd;      p.cBi = Vv;
    p.beta = 1.f;
    p.Cin = mean; p.ciRow = Vv; p.ciCol = 1; p.ciBo = Hh * Ll * Vv; p.ciBi = Ll * Vv;
    p.Kdim = Ll;
    launch_gemm(stream, p, Ll, Vv, BH);
  }
  // ---- 14) output projection: out = smp @ Wo^T + bo
  {
    GemmP p = mk();
    p.A = smp; p.aRow = Dd; p.aCol = 1;  p.aBo = Ll * Dd;
    p.B = Wo;  p.bRow = 1;  p.bCol = Dd; p.bBo = 0;
    p.C = out; p.cRow = Dd; p.cCol = 1;  p.cBo = Ll * Dd;
    p.bias = bo; p.biasMode = 1;
    p.Kdim = Dd;
    launch_gemm(stream, p, Ll, Dd, Bb);
  }

  // ---- 15) log-joint traces
  gemmMM(Kin2, Cqm, Tq,  1.f, 0.f, nullptr);
  gemmMM(Tq,  Kin2, Aqm, 1.f, 0.f, nullptr);
  gemmMM(Kin2, Ckm, Tk,  1.f, 0.f, nullptr);
  gemmMM(Tk,  Kin2, Akm, 1.f, 0.f, nullptr);
  float* lj = out + (size_t)Bb * Ll * Dd;
  set_scalar_kernel<<<1, 1, 0, stream>>>(lj, 0.f);
  logjoint_kernel<<<BH, 256, 0, stream>>>(Aqm, Akm, T0, Kin2, Cqm, Ckm,
                                          lj, inv_n4, 1.0f / BH);
}